// FeedForwardNetwork_16183436771994
// MI455X (gfx1250) — compile-verified
//
#include <hip/hip_runtime.h>

typedef float v2f __attribute__((ext_vector_type(2)));
typedef float v8f __attribute__((ext_vector_type(8)));

// D(16x16,f32) = A(16x4,f32) x B(4x16,f32) + C  -- CDNA5 V_WMMA_F32_16X16X4_F32
__device__ __forceinline__ v8f wmma_f32(v2f a, v2f b, v8f c) {
  return __builtin_amdgcn_wmma_f32_16x16x4_f32(
      /*neg_a=*/false, a, /*neg_b=*/false, b,
      /*c_mod=*/(short)0, c, /*reuse_a=*/false, /*reuse_b=*/false);
}

// fast sigmoid: v_exp_f32 + v_rcp_f32 (avoids IEEE div_scale/div_fmas sequence)
__device__ __forceinline__ float fast_sigmoid(float v) {
  return __builtin_amdgcn_rcpf(1.0f + __expf(-v));
}

// B fragment for k-chunk `kc` (rows 4*kc..4*kc+3) and 16-wide column tile `jt`
// from a row-major (K x ncols) weight matrix.
// Layout: lanes 0-15 rows {4kc+0,4kc+1}, lanes 16-31 rows {4kc+2,4kc+3};
// lane's column = jt*16 + (lane&15).
__device__ __forceinline__ v2f load_b_frag(const float* __restrict__ W, int ncols,
                                           int kc, int jt, int m, int kr) {
  const float* p = W + (size_t)(kc * 4 + kr) * ncols + jt * 16 + m;
  v2f r;
  r.x = p[0];
  r.y = p[ncols];
  return r;
}

__global__ __launch_bounds__(128) void equi_ffn_kernel(
    const float* __restrict__ xin,  const float* __restrict__ attr,
    const float* __restrict__ W1_0, const float* __restrict__ W1_1,
    const float* __restrict__ W1_2, const float* __restrict__ b1,
    const float* __restrict__ W2_0, const float* __restrict__ W2_1,
    const float* __restrict__ W2_2, const float* __restrict__ b2,
    float* __restrict__ out, int Nn) {
  // LDS: 16x672 activated stage-1 output (cols 0..383 silu scalars = GEMM2-l0 A,
  //      cols 384..671 sigmoid gates) + 16x192 scratch. scr doubles as the
  //      staging buffer for the x0 (16x128) tile during phase 1 (no spills).
  __shared__ __align__(16) float h0a[16 * 672];   // 43008 B
  __shared__ __align__(16) float scr[16 * 192];   // 12288 B
  __shared__ float attr_s[16];

  const int tid  = threadIdx.x;
  const int lane = tid & 31;
  const int wv   = tid >> 5;       // wave 0..3
  const int m    = lane & 15;      // row (A) / column (B,C,D) within 16
  const int hi   = lane >> 4;      // lane-half select
  const int kh   = hi * 2;         // K sub-offset within a 4-chunk
  const int n0   = blockIdx.x * 16;

  int rowm = n0 + m; if (rowm >= Nn) rowm = Nn - 1;   // clamped load row
  const float* xrow = xin + (size_t)rowm * 480;

  if (tid < 16) {
    int rr = n0 + tid; if (rr >= Nn) rr = Nn - 1;
    attr_s[tid] = attr[rr];
  }
  // Stage x0 tile (16 rows x 128 cols) into scr: thread t copies 16 floats of
  // row t>>3 starting at col (t&7)*16. 128-bit global loads / LDS stores.
  {
    const int r  = tid >> 3;
    const int c0 = (tid & 7) * 16;
    int rr = n0 + r; if (rr >= Nn) rr = Nn - 1;
    const float* src = xin + (size_t)rr * 480 + c0;
    float4 v0 = *(const float4*)(src + 0);
    float4 v1 = *(const float4*)(src + 4);
    float4 v2 = *(const float4*)(src + 8);
    float4 v3 = *(const float4*)(src + 12);
    *(float4*)(&scr[r * 128 + c0 + 0])  = v0;
    *(float4*)(&scr[r * 128 + c0 + 4])  = v1;
    *(float4*)(&scr[r * 128 + c0 + 8])  = v2;
    *(float4*)(&scr[r * 128 + c0 + 12]) = v3;
  }
  __syncthreads();

  const float inv128 = 0.088388347648318447f;  // 1/sqrt(128)
  const float inv64  = 0.125f;                 // 1/sqrt(64)
  const float inv32  = 0.17677669529663687f;   // 1/sqrt(32)
  const float inv384 = 0.051031036307982884f;  // 1/sqrt(384)
  const float inv192 = 0.072168783648703216f;  // 1/sqrt(192)
  const float inv96  = 0.10206207261596575f;   // 1/sqrt(96)

  // ---------------- Phase 1: GEMM1 l=0  (16x128)x(128x672) + b1, activate ----
  for (int j = wv; j < 42; j += 4) {            // 672/16 column tiles, by wave
    v8f acc = {};
    #pragma unroll 8
    for (int c = 0; c < 32; ++c) {
      v2f a = *(const v2f*)(&scr[m * 128 + c * 4 + kh]);   // ds_load_b64
      acc = wmma_f32(a, load_b_frag(W1_0, 672, c, j, m, kh), acc);
    }
    const int o = j * 16 + m;
    const float bias = b1[o];
    const bool scal = (j < 24);                 // cols <384: silu; else gate
    #pragma unroll
    for (int r = 0; r < 8; ++r) {
      const int mrow = r + hi * 8;
      float v = acc[r] * inv128 * attr_s[mrow] + bias;
      float sg = fast_sigmoid(v);
      h0a[mrow * 672 + o] = scal ? v * sg : sg;
    }
  }
  __syncthreads();   // h0a complete; scr (x0 copy) now dead

  // ---------------- Phase 2: GEMM2 l=0  (16x384)x(384x128) + b2 --------------
  for (int j = wv; j < 8; j += 4) {
    v8f acc = {};
    #pragma unroll 8
    for (int c = 0; c < 96; ++c) {
      v2f a = *(const v2f*)(&h0a[m * 672 + c * 4 + kh]);   // silu scalars as A
      acc = wmma_f32(a, load_b_frag(W2_0, 128, c, j, m, kh), acc);
    }
    const int o = j * 16 + m;
    const float bias = b2[o];
    #pragma unroll
    for (int r = 0; r < 8; ++r) {
      const int mrow = r + hi * 8;
      const int row = n0 + mrow;
      float v = acc[r] * inv384 * attr_s[mrow] + bias;
      if (row < Nn) out[(size_t)row * 480 + o] = v;
    }
  }

  // ---------------- Phase 3: l=1, per component i (3x) -----------------------
  for (int i = 0; i < 3; ++i) {
    v2f a1[16];                                  // A = x1[:,:,i] (16x64)
    #pragma unroll
    for (int c = 0; c < 16; ++c) {
      const int mc = c * 4 + kh;
      const float* p = xrow + 128 + i;           // col = 128 + m*3 + i
      v2f t; t.x = p[mc * 3]; t.y = p[(mc + 1) * 3];
      a1[c] = t;
    }
    for (int j = wv; j < 12; j += 4) {           // GEMM1: (16x64)x(64x192)
      v8f acc = {};
      #pragma unroll
      for (int c = 0; c < 16; ++c)
        acc = wmma_f32(a1[c], load_b_frag(W1_1, 192, c, j, m, kh), acc);
      const int o = j * 16 + m;
      #pragma unroll
      for (int r = 0; r < 8; ++r) {
        const int mrow = r + hi * 8;
        const float g = h0a[mrow * 672 + 384 + o];       // sigmoid gate g1
        scr[mrow * 192 + o] = acc[r] * inv64 * attr_s[mrow] * g;
      }
    }
    __syncthreads();
    {                                            // GEMM2: (16x192)x(192x64)
      const int j2 = wv;                         // 4 column tiles, 1 per wave
      v8f acc = {};
      #pragma unroll 8
      for (int c = 0; c < 48; ++c) {
        v2f a = *(const v2f*)(&scr[m * 192 + c * 4 + kh]);
        acc = wmma_f32(a, load_b_frag(W2_1, 64, c, j2, m, kh), acc);
      }
      const int o = j2 * 16 + m;
      #pragma unroll
      for (int r = 0; r < 8; ++r) {
        const int mrow = r + hi * 8;
        const int row = n0 + mrow;
        float v = acc[r] * inv192 * attr_s[mrow];
        if (row < Nn) out[(size_t)row * 480 + 128 + o * 3 + i] = v;
      }
    }
    __syncthreads();
  }

  // ---------------- Phase 4: l=2, per component i (5x) -----------------------
  for (int i = 0; i < 5; ++i) {
    v2f a2[8];                                   // A = x2[:,:,i] (16x32)
    #pragma unroll
    for (int c = 0; c < 8; ++c) {
      const int mc = c * 4 + kh;
      const float* p = xrow + 320 + i;           // col = 320 + m*5 + i
      v2f t; t.x = p[mc * 5]; t.y = p[(mc + 1) * 5];
      a2[c] = t;
    }
    for (int j = wv; j < 6; j += 4) {            // GEMM1: (16x32)x(32x96)
      v8f acc = {};
      #pragma unroll
      for (int c = 0; c < 8; ++c)
        acc = wmma_f32(a2[c], load_b_frag(W1_2, 96, c, j, m, kh), acc);
      const int o = j * 16 + m;
      #pragma unroll
      for (int r = 0; r < 8; ++r) {
        const int mrow = r + hi * 8;
        const float g = h0a[mrow * 672 + 576 + o];       // sigmoid gate g2
        scr[mrow * 96 + o] = acc[r] * inv32 * attr_s[mrow] * g;
      }
    }
    __syncthreads();
    if (wv < 2) {                                // GEMM2: (16x96)x(96x32)
      const int j2 = wv;                         // 2 column tiles
      v8f acc = {};
      #pragma unroll
      for (int c = 0; c < 24; ++c) {
        v2f a = *(const v2f*)(&scr[m * 96 + c * 4 + kh]);
        acc = wmma_f32(a, load_b_frag(W2_2, 32, c, j2, m, kh), acc);
      }
      const int o = j2 * 16 + m;
      #pragma unroll
      for (int r = 0; r < 8; ++r) {
        const int mrow = r + hi * 8;
        const int row = n0 + mrow;
        float v = acc[r] * inv96 * attr_s[mrow];
        if (row < Nn) out[(size_t)row * 480 + 320 + o * 5 + i] = v;
      }
    }
    __syncthreads();
  }
}

extern "C" void kernel_launch(void* const* d_in, const int* in_sizes, int n_in,
                              void* d_out, int out_size, void* d_ws, size_t ws_size,
                              hipStream_t stream) {
  const float* node_input = (const float*)d_in[0];
  const float* node_attr  = (const float*)d_in[1];
  const float* W1_0 = (const float*)d_in[2];
  const float* W1_1 = (const float*)d_in[3];
  const float* W1_2 = (const float*)d_in[4];
  const float* b1   = (const float*)d_in[5];
  const float* W2_0 = (const float*)d_in[6];
  const float* W2_1 = (const float*)d_in[7];
  const float* W2_2 = (const float*)d_in[8];
  const float* b2   = (const float*)d_in[9];
  float* out = (float*)d_out;

  const int Nn = in_sizes[0] / 480;
  const int blocks = (Nn + 15) / 16;
  equi_ffn_kernel<<<blocks, 128, 0, stream>>>(node_input, node_attr,
                                              W1_0, W1_1, W1_2, b1,
                                              W2_0, W2_1, W2_2, b2,
                                              out, Nn);
}